// GraphConv_20864951124336
// MI455X (gfx1250) — compile-verified
//
#include <hip/hip_runtime.h>

typedef float v2f __attribute__((ext_vector_type(2)));
typedef float v8f __attribute__((ext_vector_type(8)));

// ---------------------------------------------------------------------------
// Kernel A: h = nfeat @ W^T + b  (fp32 WMMA 16x16x4), fused epilogue:
//   h -> workspace (for edge gather)
//   d_out = relu(h + root_emb) / degs   (root term, init for scatter-add)
// One wave = one 16x16 output tile. 8 waves/block cover all 8 N-tiles (F=128)
// of one 16-row M-block, so W tiles are reused through L1/L2.
// ---------------------------------------------------------------------------
__global__ __launch_bounds__(256)
void node_linear_wmma(const float* __restrict__ nfeat,   // [N,128]
                      const float* __restrict__ W,       // [128,128] row-major
                      const float* __restrict__ b,       // [128]
                      const float* __restrict__ root_emb,// [128]
                      const float* __restrict__ degs,    // [N]
                      float* __restrict__ h,             // [N,128] workspace
                      float* __restrict__ out,           // [N,128] output
                      int N)
{
    const int lane   = threadIdx.x & 31;
    const int wave   = threadIdx.x >> 5;        // 0..7 -> N-tile index
    const int tile_m = blockIdx.x * 16;
    const int tile_n = wave * 16;
    if (tile_m >= N) return;                    // block-uniform

    const int half = lane >> 4;                 // 0: K={0,1}, 1: K={2,3}
    const int l16  = lane & 15;

    // A: lane holds row (tile_m + l16), two consecutive K values
    int arow = tile_m + l16;
    if (arow >= N) arow = N - 1;                // clamp (N%16==0 in practice)
    const float* aptr = nfeat + (size_t)arow * 128 + 2 * half;
    // B[k][n] = W[n*128 + k]: lane holds col (tile_n + l16), two consecutive K
    const float* bptr = W + (size_t)(tile_n + l16) * 128 + 2 * half;

    v8f c = {};
    #pragma unroll
    for (int k0 = 0; k0 < 128; k0 += 4) {
        v2f a  = *(const v2f*)(aptr + k0);
        v2f bb = *(const v2f*)(bptr + k0);
        // D = A*B + C, fp32, emits v_wmma_f32_16x16x4_f32
        c = __builtin_amdgcn_wmma_f32_16x16x4_f32(
                /*neg_a=*/false, a, /*neg_b=*/false, bb,
                /*c_mod=*/(short)0, c, /*reuse_a=*/false, /*reuse_b=*/false);
    }

    const int   col  = tile_n + l16;
    const float bias = b[col];
    const float re   = root_emb[col];

    #pragma unroll
    for (int v = 0; v < 8; ++v) {
        const int row = tile_m + v + 8 * half;  // C/D layout: VGPR v -> row
        if (row < N) {
            const size_t idx = (size_t)row * 128 + col;
            const float hv = c[v] + bias;
            h[idx] = hv;
            float r = hv + re;
            r = r > 0.0f ? r : 0.0f;
            out[idx] = r / degs[row];
        }
    }
}

// ---------------------------------------------------------------------------
// Kernel B: per-edge  e = norm * relu(h[src] + efeat@We^T + be)
//           out[dst] += e   (hardware global_atomic_add_f32)
// One wave per edge; lane handles 4 consecutive features (128 = 32*4).
// We (128x7) + be staged in LDS.
// ---------------------------------------------------------------------------
__global__ __launch_bounds__(256)
void edge_scatter(const float* __restrict__ efeat, // [E,7]
                  const float* __restrict__ norm,  // [E]
                  const int*   __restrict__ src,   // [E]
                  const int*   __restrict__ dst,   // [E]
                  const float* __restrict__ We,    // [128,7]
                  const float* __restrict__ be,    // [128]
                  const float* __restrict__ h,     // [N,128]
                  float* __restrict__ out,         // [N,128]
                  int E)
{
    __shared__ float sWe[128 * 7];
    __shared__ float sbe[128];
    for (int i = threadIdx.x; i < 128 * 7; i += 256) sWe[i] = We[i];
    if (threadIdx.x < 128) sbe[threadIdx.x] = be[threadIdx.x];
    __syncthreads();

    const int lane = threadIdx.x & 31;
    const int wave = threadIdx.x >> 5;
    const int e    = blockIdx.x * 8 + wave;
    if (e >= E) return;                          // wave-uniform

    const int   s   = src[e];
    const int   d   = dst[e];
    const float nrm = norm[e];

    float ef[7];
    const float* ep = efeat + (size_t)e * 7;
    #pragma unroll
    for (int k = 0; k < 7; ++k) ef[k] = ep[k];

    const int f0 = lane * 4;
    const float4 hs = *(const float4*)(h + (size_t)s * 128 + f0);
    const float hv[4] = { hs.x, hs.y, hs.z, hs.w };

    float* op = out + (size_t)d * 128 + f0;
    #pragma unroll
    for (int j = 0; j < 4; ++j) {
        const int f = f0 + j;
        float acc = sbe[f];
        const float* wrow = &sWe[f * 7];
        #pragma unroll
        for (int k = 0; k < 7; ++k) acc += ef[k] * wrow[k];
        float v = hv[j] + acc;
        v = v > 0.0f ? v : 0.0f;
        unsafeAtomicAdd(op + j, v * nrm);        // global_atomic_add_f32, no return
    }
}

// ---------------------------------------------------------------------------
extern "C" void kernel_launch(void* const* d_in, const int* in_sizes, int n_in,
                              void* d_out, int out_size, void* d_ws, size_t ws_size,
                              hipStream_t stream) {
    const float* nfeat = (const float*)d_in[0];
    const float* efeat = (const float*)d_in[1];
    const float* degs  = (const float*)d_in[2];
    const float* norm  = (const float*)d_in[3];
    const int*   src   = (const int*)  d_in[4];
    const int*   dst   = (const int*)  d_in[5];
    const float* W     = (const float*)d_in[6];
    const float* b     = (const float*)d_in[7];
    const float* We    = (const float*)d_in[8];
    const float* be    = (const float*)d_in[9];
    const float* root  = (const float*)d_in[10];

    const int N = in_sizes[2];   // degs: [N]
    const int E = in_sizes[4];   // src:  [E]

    float* h   = (float*)d_ws;   // N*128 floats (~51 MB, fits in L2)
    float* out = (float*)d_out;

    // Phase 1: node linear + root term (initializes d_out fully)
    node_linear_wmma<<<dim3((N + 15) / 16), dim3(256), 0, stream>>>(
        nfeat, W, b, root, degs, h, out, N);

    // Phase 2: edge encode + gather + scatter-add
    edge_scatter<<<dim3((E + 7) / 8), dim3(256), 0, stream>>>(
        efeat, norm, src, dst, We, be, h, out, E);
}